// LightGCN_56246891708625
// MI455X (gfx1250) — compile-verified
//
#include <hip/hip_runtime.h>
#include <hip/hip_bf16.h>

#define NUM_USERS 100000
#define NUM_ITEMS 50000
#define EMB_DIM   64
#define NNZ       4800000
#define N_LAYERS  3
#define BATCH     16384
#define NTOT      (NUM_USERS + NUM_ITEMS)

typedef __attribute__((ext_vector_type(2))) float v2f;
typedef __attribute__((ext_vector_type(8))) float v8f;

// ---------------------------------------------------------------------------
// init: acc = cur = concat(user_table, item_table); next = 0; out = 0
// B128 vectorized (each float4 lies within one 64-wide row).
// ---------------------------------------------------------------------------
__global__ void lgcn_init(const float* __restrict__ ut, const float* __restrict__ it,
                          float* __restrict__ acc, float* __restrict__ cur,
                          float* __restrict__ nxt, float* __restrict__ out) {
    const int total4 = NTOT * EMB_DIM / 4;           // 2,400,000
    int i = blockIdx.x * blockDim.x + threadIdx.x;
    if (i < total4) {
        int row = i >> 4;                            // (4*i)/64
        float4 v;
        if (row < NUM_USERS) v = ((const float4*)ut)[i];
        else                 v = ((const float4*)it)[i - NUM_USERS * (EMB_DIM / 4)];
        ((float4*)acc)[i] = v;
        ((float4*)cur)[i] = v;
        ((float4*)nxt)[i] = make_float4(0.f, 0.f, 0.f, 0.f);
    }
    if (i == 0) out[0] = 0.f;
}

// ---------------------------------------------------------------------------
// scatter: next[row] += val * cur[col], one wave per EDGES_PER_WAVE edges.
// Index loads are wave-uniform (scalarize); per lane: float2 gather from the
// source row (L2-resident, 38.4MB table < 192MB L2) + two f32 atomics.
// ---------------------------------------------------------------------------
#define EDGES_PER_WAVE 8

__global__ void lgcn_scatter(const float* __restrict__ val,
                             const int*   __restrict__ rowi,
                             const int*   __restrict__ coli,
                             const float* __restrict__ cur,
                             float*       __restrict__ nxt) {
    const int lane = threadIdx.x & 31;
    const int wave = (blockIdx.x * blockDim.x + threadIdx.x) >> 5;
    const int e0 = wave * EDGES_PER_WAVE;
#pragma unroll
    for (int k = 0; k < EDGES_PER_WAVE; ++k) {
        const int e = e0 + k;
        if (e >= NNZ) return;
        if (e + 512 < NNZ) {                  // stream the edge lists ahead
            __builtin_prefetch(&coli[e + 512], 0, 0);
            __builtin_prefetch(&rowi[e + 512], 0, 0);
            __builtin_prefetch(&val [e + 512], 0, 0);
        }
        const int   c = coli[e];
        const int   r = rowi[e];
        const float a = val[e];
        const float2 src = *(const float2*)(cur + (size_t)c * EMB_DIM + 2 * lane);
        float* dst = nxt + (size_t)r * EMB_DIM + 2 * lane;
        atomicAdd(dst,     a * src.x);
        atomicAdd(dst + 1, a * src.y);
    }
}

// ---------------------------------------------------------------------------
// add_zero: acc += next;  zero the retiring buffer (becomes next-next)
// ---------------------------------------------------------------------------
__global__ void lgcn_add_zero(float* __restrict__ acc, const float* __restrict__ nxt,
                              float* __restrict__ retire) {
    const int total4 = NTOT * EMB_DIM / 4;
    int i = blockIdx.x * blockDim.x + threadIdx.x;
    if (i >= total4) return;
    float4 a = ((float4*)acc)[i];
    float4 n = ((const float4*)nxt)[i];
    a.x += n.x; a.y += n.y; a.z += n.z; a.w += n.w;
    ((float4*)acc)[i] = a;
    ((float4*)retire)[i] = make_float4(0.f, 0.f, 0.f, 0.f);
}

// ---------------------------------------------------------------------------
// loss: one wave handles 16 batch triples. x = diag( U * (P-N)^T ) via 16
// chained V_WMMA_F32_16X16X4_F32 (exact fp32). A(16x4) and B(4x16) share the
// same per-lane indexing: row = lane&15, k = 4*chunk + 2*(lane>>4) + {0,1}.
// mean_emb = acc/4 folded in as x *= 1/16. Stable softplus(-x), wave reduce,
// one atomicAdd per wave scaled by 1/BATCH.
// ---------------------------------------------------------------------------
__global__ void lgcn_loss(const float* __restrict__ acc,
                          const int* __restrict__ uidx,
                          const int* __restrict__ pidx,
                          const int* __restrict__ nidx,
                          float* __restrict__ out) {
    const int lane = threadIdx.x & 31;
    const int wave = (blockIdx.x * blockDim.x + threadIdx.x) >> 5;  // group of 16
    const int r    = lane & 15;
    const int half = lane >> 4;
    const int b    = wave * 16 + r;

    const float* uRow = acc + (size_t)uidx[b] * EMB_DIM;
    const float* pRow = acc + (size_t)(NUM_USERS + pidx[b]) * EMB_DIM;
    const float* nRow = acc + (size_t)(NUM_USERS + nidx[b]) * EMB_DIM;

    v8f c = {};
#pragma unroll
    for (int kk = 0; kk < 16; ++kk) {
        const int koff = kk * 4 + half * 2;
        const float2 uv = *(const float2*)(uRow + koff);
        const float2 pv = *(const float2*)(pRow + koff);
        const float2 nv = *(const float2*)(nRow + koff);
        v2f A; A.x = uv.x;        A.y = uv.y;
        v2f B; B.x = pv.x - nv.x; B.y = pv.y - nv.y;
        c = __builtin_amdgcn_wmma_f32_16x16x4_f32(
                /*neg_a=*/false, A, /*neg_b=*/false, B,
                /*c_mod=*/(short)0, c, /*reuse_a=*/false, /*reuse_b=*/false);
    }

    // Diagonal of C: (d,d) for d<8 lives in cmat[d] of lane d;
    // for d>=8 it lives in cmat[d-8] of lane d+16 (= 24 + (d-8)).
    float x = 0.f;
#pragma unroll
    for (int rr = 0; rr < 8; ++rr) {
        if (lane == rr)      x = c[rr];
        if (lane == 24 + rr) x = c[rr];
    }
    const bool valid = (lane < 8) || (lane >= 24);

    x *= 0.0625f;                       // (1/4 * 1/4) from mean_emb scaling
    const float t  = -x;                // -log_sigmoid(x) = softplus(-x)
    float li = valid ? (fmaxf(t, 0.f) + log1pf(expf(-fabsf(t)))) : 0.f;

#pragma unroll
    for (int off = 16; off > 0; off >>= 1)
        li += __shfl_xor(li, off, 32);
    if (lane == 0)
        atomicAdd(out, li * (1.0f / BATCH));
}

// ---------------------------------------------------------------------------
extern "C" void kernel_launch(void* const* d_in, const int* in_sizes, int n_in,
                              void* d_out, int out_size, void* d_ws, size_t ws_size,
                              hipStream_t stream) {
    const float* ut      = (const float*)d_in[0];
    const float* it      = (const float*)d_in[1];
    const float* adj_val = (const float*)d_in[2];
    const int*   adj_row = (const int*)  d_in[3];
    const int*   adj_col = (const int*)  d_in[4];
    const int*   uidx    = (const int*)  d_in[5];
    const int*   pidx    = (const int*)  d_in[6];
    const int*   nidx    = (const int*)  d_in[7];
    float* out = (float*)d_out;

    const size_t tblElems = (size_t)NTOT * EMB_DIM;   // 9.6M floats = 38.4MB
    float* acc  = (float*)d_ws;
    float* bufA = acc  + tblElems;
    float* bufB = bufA + tblElems;

    const int total4      = NTOT * EMB_DIM / 4;
    const int elemBlocks  = (total4 + 255) / 256;
    const int scatBlocks  = (NNZ + EDGES_PER_WAVE * 8 - 1) / (EDGES_PER_WAVE * 8); // 8 waves/block
    const int lossBlocks  = (BATCH / 16) / 8;         // 1024 waves / 8 waves-per-block

    lgcn_init<<<elemBlocks, 256, 0, stream>>>(ut, it, acc, bufA, bufB, out);

    float* cur = bufA;
    float* nxt = bufB;
    for (int layer = 0; layer < N_LAYERS; ++layer) {
        lgcn_scatter<<<scatBlocks, 256, 0, stream>>>(adj_val, adj_row, adj_col, cur, nxt);
        lgcn_add_zero<<<elemBlocks, 256, 0, stream>>>(acc, nxt, cur);
        float* t = cur; cur = nxt; nxt = t;
    }

    lgcn_loss<<<lossBlocks, 256, 0, stream>>>(acc, uidx, pidx, nidx, out);
}